// ExpertD_30769145709060
// MI455X (gfx1250) — compile-verified
//
#include <hip/hip_runtime.h>

typedef __attribute__((ext_vector_type(2))) float v2f;
typedef __attribute__((ext_vector_type(8))) float v8f;

#define HID 32
#define NGRAPH 1024
#define BN_EPS 1e-5f

// ---------------- init: deg=1 (self loop), pool/cnt = 0 ----------------
__global__ void k_init(float* __restrict__ deg, float* __restrict__ pool,
                       float* __restrict__ cnt, int n_nodes) {
  int i = blockIdx.x * blockDim.x + threadIdx.x;
  if (i < n_nodes) deg[i] = 1.0f;
  if (i < NGRAPH * HID) pool[i] = 0.0f;
  if (i < NGRAPH) cnt[i] = 0.0f;
}

// ---------------- degree scatter: deg[dst] += 1 ----------------
__global__ void k_deg(const long long* __restrict__ dst, float* __restrict__ deg,
                      int n_edges) {
  int e = blockIdx.x * blockDim.x + threadIdx.x;
  if (e >= n_edges) return;
  atomicAdd(&deg[(int)dst[e]], 1.0f);
}

// ---------------- dinv = rsqrt(deg); s1 init with self-loop ----------------
__global__ void k_dinv_s1(const float* __restrict__ x, float* dinv_io,
                          float* __restrict__ s1, int n_nodes) {
  int n = blockIdx.x * blockDim.x + threadIdx.x;
  if (n >= n_nodes) return;
  float d = rsqrtf(dinv_io[n]);   // deg >= 1 always (self-loop added)
  dinv_io[n] = d;
  s1[n] = x[n] * d * d;           // self-loop: norm = dinv^2, h = x (IN_DIM==1 folded)
}

// ---------------- layer-1 scalar edge scatter ----------------
__global__ void k_s1_scatter(const long long* __restrict__ src,
                             const long long* __restrict__ dst,
                             const float* __restrict__ x,
                             const float* __restrict__ dinv,
                             float* __restrict__ s1, int n_edges) {
  int e = blockIdx.x * blockDim.x + threadIdx.x;
  if (e >= n_edges) return;
  int s = (int)src[e], d = (int)dst[e];
  atomicAdd(&s1[d], x[s] * dinv[s] * dinv[d]);
}

// ---- layer-1 transform (outer product, VALU) + init agg2 with self-loop ----
__global__ void k_transform1(const float* __restrict__ s1, const float* __restrict__ dinv,
                             const float* __restrict__ W1, const float* __restrict__ b1,
                             const float* __restrict__ g, const float* __restrict__ bb,
                             const float* __restrict__ bm, const float* __restrict__ bv,
                             float* __restrict__ z1, float* __restrict__ agg2,
                             int n_nodes) {
  long long t = (long long)blockIdx.x * blockDim.x + threadIdx.x;
  if (t >= (long long)n_nodes * HID) return;
  int n = (int)(t >> 5);
  int j = (int)(t & 31);
  float val = s1[n] * W1[j] + b1[j];
  float sc = g[j] * rsqrtf(bv[j] + BN_EPS);
  val = (val - bm[j]) * sc + bb[j];
  val = fmaxf(val, 0.0f);
  z1[t] = val;
  float dn = dinv[n];
  agg2[t] = dn * dn * val;        // self-loop contribution for next aggregation
}

// ---------------- 32-wide edge scatter: agg[dst] += norm * z[src] ----------------
__global__ void k_edge_scatter32(const long long* __restrict__ src,
                                 const long long* __restrict__ dst,
                                 const float* __restrict__ dinv,
                                 const float* __restrict__ z,
                                 float* __restrict__ agg, int n_edges) {
  long long t = (long long)blockIdx.x * blockDim.x + threadIdx.x;
  long long e = t >> 5;
  int f = threadIdx.x & 31;
  if (e >= n_edges) return;
  int s = (int)src[e], d = (int)dst[e];   // wave-uniform (scalarized)
  float nrm = dinv[s] * dinv[d];
  float val = z[(long long)s * HID + f] * nrm;
  atomicAdd(&agg[(long long)d * HID + f], val);
}

// ---------------- WMMA node transform: out = A @ W + bias (+BN+ReLU) ----------------
// A: [n_nodes, 32] fp32, W: [32, 32] row-major. One wave handles 16 rows.
// V_WMMA_F32_16X16X4_F32: A 16x4 (2 VGPR), B 4x16 (2 VGPR), C/D 16x16 f32 (8 VGPR).
// mode 0: Zout = relu(bn(val)), AggNext = dinv^2 * Zout   (in-place on A allowed)
// mode 1: pool[batch[row]] += val  (global mean-pool numerator, layer 3)
__global__ void k_transform_wmma(const float* A, const float* __restrict__ W,
                                 const float* __restrict__ bias,
                                 const float* __restrict__ g, const float* __restrict__ bb,
                                 const float* __restrict__ bm, const float* __restrict__ bv,
                                 const float* __restrict__ dinv,
                                 float* Zout, float* AggNext,
                                 float* __restrict__ pool,
                                 const long long* __restrict__ batch,
                                 int n_nodes, int mode) {
  int wave = (blockIdx.x * blockDim.x + threadIdx.x) >> 5;
  int lane = threadIdx.x & 31;
  int row0 = wave << 4;
  if (row0 >= n_nodes) return;      // whole-wave exit; EXEC all-ones for WMMA

  int half = lane >> 4;             // 0: K pair {0,1}; 1: K pair {2,3}
  int l16  = lane & 15;

  v8f acc0 = {};                    // output cols 0..15
  v8f acc1 = {};                    // output cols 16..31

  const float* arow = A + (long long)(row0 + l16) * HID + 2 * half;
#pragma unroll
  for (int kc = 0; kc < 8; ++kc) {
    v2f a = *(const v2f*)(arow + kc * 4);             // A[row][4kc+2h .. +1]
    int k0 = kc * 4 + 2 * half;
    v2f bL = { W[k0 * HID + l16],      W[(k0 + 1) * HID + l16]      };
    v2f bH = { W[k0 * HID + 16 + l16], W[(k0 + 1) * HID + 16 + l16] };
    acc0 = __builtin_amdgcn_wmma_f32_16x16x4_f32(false, a, false, bL,
                                                 (short)0, acc0, false, false);
    acc1 = __builtin_amdgcn_wmma_f32_16x16x4_f32(false, a, false, bH,
                                                 (short)0, acc1, false, false);
  }

  int c0 = l16, c1 = 16 + l16;
  if (mode == 0) {
    float sc0 = g[c0] * rsqrtf(bv[c0] + BN_EPS);
    float sc1 = g[c1] * rsqrtf(bv[c1] + BN_EPS);
    float off0 = (bias[c0] - bm[c0]) * sc0 + bb[c0];
    float off1 = (bias[c1] - bm[c1]) * sc1 + bb[c1];
#pragma unroll
    for (int v = 0; v < 8; ++v) {
      int m = row0 + v + 8 * half;                    // C/D layout: VGPR v -> row v (+8 hi half)
      float dn = dinv[m]; float dn2 = dn * dn;
      float z0 = fmaxf(acc0[v] * sc0 + off0, 0.0f);
      float z1 = fmaxf(acc1[v] * sc1 + off1, 0.0f);
      long long base = (long long)m * HID;
      Zout[base + c0] = z0;
      Zout[base + c1] = z1;
      AggNext[base + c0] = dn2 * z0;                  // self-loop for next layer
      AggNext[base + c1] = dn2 * z1;
    }
  } else {
    float bi0 = bias[c0], bi1 = bias[c1];
#pragma unroll
    for (int v = 0; v < 8; ++v) {
      int m = row0 + v + 8 * half;
      int gi = (int)batch[m];
      atomicAdd(&pool[gi * HID + c0], acc0[v] + bi0);
      atomicAdd(&pool[gi * HID + c1], acc1[v] + bi1);
    }
  }
}

// ---------------- per-graph node counts ----------------
__global__ void k_cnt(const long long* __restrict__ batch, float* __restrict__ cnt,
                      int n_nodes) {
  int n = blockIdx.x * blockDim.x + threadIdx.x;
  if (n >= n_nodes) return;
  atomicAdd(&cnt[(int)batch[n]], 1.0f);
}

// ---------------- finalize mean pool ----------------
__global__ void k_final(const float* __restrict__ pool, const float* __restrict__ cnt,
                        float* __restrict__ out) {
  int i = blockIdx.x * blockDim.x + threadIdx.x;
  if (i >= NGRAPH * HID) return;
  out[i] = pool[i] / fmaxf(cnt[i >> 5], 1.0f);
}

extern "C" void kernel_launch(void* const* d_in, const int* in_sizes, int n_in,
                              void* d_out, int out_size, void* d_ws, size_t ws_size,
                              hipStream_t stream) {
  const float*     x     = (const float*)d_in[0];
  const long long* ei    = (const long long*)d_in[1];   // [2, E] int64
  const long long* batch = (const long long*)d_in[2];
  const float* W1 = (const float*)d_in[3];
  const float* b1 = (const float*)d_in[4];
  const float* g1 = (const float*)d_in[5];
  const float* B1 = (const float*)d_in[6];
  const float* m1 = (const float*)d_in[7];
  const float* v1 = (const float*)d_in[8];
  const float* W2 = (const float*)d_in[9];
  const float* b2 = (const float*)d_in[10];
  const float* g2 = (const float*)d_in[11];
  const float* B2 = (const float*)d_in[12];
  const float* m2 = (const float*)d_in[13];
  const float* v2 = (const float*)d_in[14];
  const float* W3 = (const float*)d_in[15];
  const float* b3 = (const float*)d_in[16];
  float* out = (float*)d_out;

  const int N = in_sizes[0];           // 100000 (multiple of 16)
  const int E = in_sizes[1] / 2;       // 3200000
  const long long* src = ei;
  const long long* dst = ei + E;

  // workspace layout (floats)
  float* ws   = (float*)d_ws;
  float* dinv = ws;                          // N   (holds deg, then dinv in place)
  float* s1   = dinv + N;                    // N
  float* buf1 = s1 + N;                      // N*32: z1, then agg3
  float* buf2 = buf1 + (size_t)N * HID;      // N*32: agg2, then z2 (in place)
  float* pool = buf2 + (size_t)N * HID;      // 1024*32
  float* cnt  = pool + NGRAPH * HID;         // 1024

  const int B = 256;
  int nb_N  = (N + B - 1) / B;
  int nb_E  = (E + B - 1) / B;
  long long nf = (long long)N * HID;
  int nb_NF = (int)((nf + B - 1) / B);
  long long ef = (long long)E * HID;
  int nb_EF = (int)((ef + B - 1) / B);
  int waves = (N + 15) / 16;
  int nb_T  = (waves * 32 + B - 1) / B;

  // norm precompute
  k_init      <<<nb_N, B, 0, stream>>>(dinv, pool, cnt, N);
  k_deg       <<<nb_E, B, 0, stream>>>(dst, dinv, E);
  k_dinv_s1   <<<nb_N, B, 0, stream>>>(x, dinv, s1, N);

  // layer 1 (scalar aggregation -> outer product + BN + ReLU)
  k_s1_scatter<<<nb_E, B, 0, stream>>>(src, dst, x, dinv, s1, E);
  k_transform1<<<nb_NF, B, 0, stream>>>(s1, dinv, W1, b1, g1, B1, m1, v1,
                                        /*z1*/buf1, /*agg2*/buf2, N);

  // layer 2: scatter z1 -> agg2, WMMA transform (in-place z2, seeds agg3 in buf1)
  k_edge_scatter32<<<nb_EF, B, 0, stream>>>(src, dst, dinv, buf1, buf2, E);
  k_transform_wmma<<<nb_T, B, 0, stream>>>(buf2, W2, b2, g2, B2, m2, v2, dinv,
                                           /*Zout*/buf2, /*AggNext*/buf1,
                                           pool, batch, N, /*mode*/0);

  // layer 3: scatter z2 -> agg3, WMMA transform pools directly
  k_edge_scatter32<<<nb_EF, B, 0, stream>>>(src, dst, dinv, buf2, buf1, E);
  k_transform_wmma<<<nb_T, B, 0, stream>>>(buf1, W3, b3, b3, b3, b3, b3, dinv,
                                           buf1, buf1, pool, batch, N, /*mode*/1);

  // mean pool
  k_cnt  <<<nb_N, B, 0, stream>>>(batch, cnt, N);
  k_final<<<(NGRAPH * HID + B - 1) / B, B, 0, stream>>>(pool, cnt, out);
}